// Decoder_5437428597075
// MI455X (gfx1250) — compile-verified
//
#include <hip/hip_runtime.h>
#include <math.h>

typedef __attribute__((ext_vector_type(2))) float v2f;
typedef __attribute__((ext_vector_type(8))) float v8f;

#define NSPHERES 64
#define NG (NSPHERES / 16)      // 4 groups of 16 spheres
#define BIGF 10000000000.0f
#define SIGMA 0.002f

// One wave (32 lanes) processes a tile of 16 rays x 64 spheres.
// b[n,s] and c[n,s] of the quadratic are computed as fp32 GEMMs with
// V_WMMA_F32_16X16X4_F32 (M=16 rays, N=16 spheres, K accumulated in chunks of 4).
//
// Operand layouts (CDNA5 ISA 7.12.2, 32-bit, wave32):
//   A 16x4: reg r, lane L -> (M = L%16, K = 2*(L/16) + r)
//   B 4x16: reg r, lane L -> (K = 2*(L/16) + r, N = L%16)
//   C/D 16x16: reg p, lane L -> (M = p + 8*(L/16), N = L%16)
__global__ __launch_bounds__(256) void ray_sphere_wmma(
    const float* __restrict__ x,   // [N, 7] : o(3), t, dir(3)
    const float* __restrict__ z,   // [64, 6]: c(3), v(3)
    float* __restrict__ out,       // [2*N]  : mu then sigma
    int nRays, int nTiles)
{
  const int lane = threadIdx.x & 31;
  const int waveInBlock = threadIdx.x >> 5;
  const int wavesPerBlock = blockDim.x >> 5;
  const int waveId = blockIdx.x * wavesPerBlock + waveInBlock;
  const int nWaves = gridDim.x * wavesPerBlock;

  const int half = lane >> 4;     // 0: lanes 0-15, 1: lanes 16-31
  const int l16  = lane & 15;

  // ---------- Per-sphere B operands (invariant over the ray-tile loop) ----------
  // This lane supplies sphere s = g*16 + l16 for group g.
  float Bb[NG][2][2];   // [group][K-chunk (k0=0,4)][reg]
  float Bc[NG][3][2];   // [group][K-chunk (k0=0,4,8)][reg]
#pragma unroll
  for (int g = 0; g < NG; ++g) {
    const float* zs = z + (g * 16 + l16) * 6;
    float cx = zs[0], cy = zs[1], cz = zs[2];
    float vx = zs[3], vy = zs[4], vz = zs[5];
    float cc  = cx * cx + cy * cy + cz * cz;
    float vv  = vx * vx + vy * vy + vz * vz;
    float cv2 = 2.0f * (cx * vx + cy * vy + cz * vz);
    // b-GEMM column (K=8):  oc terms dotted with d
    float colb[8]  = { cx, cy, cz, vx, vy, vz, 1.0f, 0.0f };
    // c-GEMM column (K=12): |oc|^2 - 1 expansion
    float colc[12] = { cc, vv, cv2, -2.0f * cx, -2.0f * cy, -2.0f * cz,
                       -2.0f * vx, -2.0f * vy, -2.0f * vz, 1.0f, 0.0f, 0.0f };
#pragma unroll
    for (int j = 0; j < 2; ++j) {
      Bb[g][j][0] = half ? colb[4 * j + 2] : colb[4 * j + 0];
      Bb[g][j][1] = half ? colb[4 * j + 3] : colb[4 * j + 1];
    }
#pragma unroll
    for (int j = 0; j < 3; ++j) {
      Bc[g][j][0] = half ? colc[4 * j + 2] : colc[4 * j + 0];
      Bc[g][j][1] = half ? colc[4 * j + 3] : colc[4 * j + 1];
    }
  }

  // ---------- Grid-stride loop over 16-ray tiles ----------
  for (int tile = waveId; tile < nTiles; tile += nWaves) {
    const int ray = tile * 16 + l16;      // both wave halves cover the same 16 rays
    const float* xr = x + (size_t)ray * 7;

    // Prefetch next tile's ray data for this wave (global_prefetch_b8).
    __builtin_prefetch(xr + (size_t)nWaves * 16 * 7, 0, 3);

    float ox = xr[0], oy = xr[1], oz = xr[2];
    float t  = xr[3];
    float dx = xr[4], dy = xr[5], dz = xr[6];

    // Fast-path normalize: inputs are O(1); raw v_sqrt/v_rcp (~1 ulp) is
    // well inside the reference's tolerance and avoids the IEEE fixup code.
    float nrm = __builtin_amdgcn_sqrtf(dx * dx + dy * dy + dz * dz);
    float inv = __builtin_amdgcn_rcpf(nrm + 1e-12f);
    dx *= inv; dy *= inv; dz *= inv;

    float d_dot_o = dx * ox + dy * oy + dz * oz;
    float oo      = ox * ox + oy * oy + oz * oz;

    // A rows for this ray
    float ab[8]  = { dx, dy, dz, t * dx, t * dy, t * dz, -d_dot_o, 0.0f };
    float ac[12] = { 1.0f, t * t, t, ox, oy, oz, t * ox, t * oy, t * oz,
                     oo - 1.0f, 0.0f, 0.0f };

    v2f Ab[2], Ac[3];
#pragma unroll
    for (int j = 0; j < 2; ++j) {
      Ab[j][0] = half ? ab[4 * j + 2] : ab[4 * j + 0];
      Ab[j][1] = half ? ab[4 * j + 3] : ab[4 * j + 1];
    }
#pragma unroll
    for (int j = 0; j < 3; ++j) {
      Ac[j][0] = half ? ac[4 * j + 2] : ac[4 * j + 0];
      Ac[j][1] = half ? ac[4 * j + 3] : ac[4 * j + 1];
    }

    float best[8];
#pragma unroll
    for (int p = 0; p < 8; ++p) best[p] = BIGF;

#pragma unroll
    for (int g = 0; g < NG; ++g) {
      v8f accB = {};   // b tile: 16 rays x 16 spheres
      v8f accC = {};   // c tile
#pragma unroll
      for (int j = 0; j < 2; ++j) {
        v2f B; B[0] = Bb[g][j][0]; B[1] = Bb[g][j][1];
        accB = __builtin_amdgcn_wmma_f32_16x16x4_f32(
            false, Ab[j], false, B, (short)0, accB, false, false);
      }
#pragma unroll
      for (int j = 0; j < 3; ++j) {
        v2f B; B[0] = Bc[g][j][0]; B[1] = Bc[g][j][1];
        accC = __builtin_amdgcn_wmma_f32_16x16x4_f32(
            false, Ac[j], false, B, (short)0, accC, false, false);
      }
      // Per-lane postprocessing: 8 (ray, sphere) pairs per lane.
#pragma unroll
      for (int p = 0; p < 8; ++p) {
        float b = accB[p];
        float c = accC[p];
        float disc = b * b - c;
        float sq = __builtin_amdgcn_sqrtf(disc > 0.0f ? disc : 1.0f);
        float t0 = b - sq;
        float t1 = b + sq;
        float th = (t0 > 0.0f) ? t0 : t1;
        bool valid = (disc >= 0.0f) && (t1 > 0.0f);
        float dist = valid ? th : BIGF;
        best[p] = fminf(best[p], dist);
      }
    }

    // Min over the 16 sphere-lanes. XOR masks < 16 never cross the half-wave,
    // so each half reduces its own 8 rays independently.
#pragma unroll
    for (int off = 8; off >= 1; off >>= 1) {
#pragma unroll
      for (int p = 0; p < 8; ++p)
        best[p] = fminf(best[p], __shfl_xor(best[p], off, 32));
    }

    if (l16 == 0) {
      const int base = tile * 16 + half * 8;   // half 0 -> rays 0..7, half 1 -> 8..15
#pragma unroll
      for (int p = 0; p < 8; ++p) {
        out[base + p]         = best[p];   // mu
        out[nRays + base + p] = SIGMA;     // sigma
      }
    }
  }
}

extern "C" void kernel_launch(void* const* d_in, const int* in_sizes, int n_in,
                              void* d_out, int out_size, void* d_ws, size_t ws_size,
                              hipStream_t stream) {
  const float* x = (const float*)d_in[0];   // (1, N, 7) fp32
  const float* z = (const float*)d_in[1];   // (1, 384)  fp32 -> 64 spheres
  float* out = (float*)d_out;               // mu (N) ++ sigma (N)

  const int nRays  = in_sizes[0] / 7;       // 1,048,576
  const int nTiles = (nRays + 15) / 16;     // 65,536

  const int threads = 256;                  // 8 waves per block
  const int blocks  = 2048;                 // 16,384 waves -> 4 tiles per wave
  ray_sphere_wmma<<<blocks, threads, 0, stream>>>(x, z, out, nRays, nTiles);
}